// Encoding_48773648614096
// MI455X (gfx1250) — compile-verified
//
#include <hip/hip_runtime.h>
#include <hip/hip_bf16.h>

// Encoding layer, fused single-pass kernel for MI455X (gfx1250, wave32).
// B=32, D=256, K=32, N=96*96=9216.
//
//   cross = Xf @ C^T            (WMMA f32 16x16x4, K-dim = D = 256)
//   A     = softmax_k(scale_k * (|x|^2 - 2*cross + |c_k|^2))
//   E     = A^T @ Xf - (sum_n A) * C   (WMMA f32 16x16x4, K-dim = n)
//
// Grid: (12 n-chunks, 32 batches). Block: 256 threads = 8 wave32.
// Each workgroup handles 768 n-rows in 8 tiles of 96, reading X once.
// LDS ~144 KB -> two workgroups co-resident per 320 KB WGP.
//
// All LDS layouts are pair-packed so every WMMA operand is one DS op:
//   XP[n/2][d][2]  (stride XPLD): stage-B B-op = b64, stage-A A-op = 2addr_b32
//   CP[d/2][k][2]               : stage-A B-op = b64
//   AP[n/2][k][2]  (stride APLD): stage-B A-op = b64

typedef float v2f __attribute__((ext_vector_type(2)));
typedef float v8f __attribute__((ext_vector_type(8)));

#define NB    9216          // H*W
#define DD    256
#define KK    32
#define TILE  96            // n-rows per tile iteration (6 waves x 16 rows in stage A)
#define ITERS 8             // tiles per workgroup
#define CHUNK 768           // n-rows per workgroup (TILE*ITERS)
#define XPLD  516           // XP row-pair stride (2*DD + 4 pad)
#define APLD  66            // AP row-pair stride (2*KK + 2 pad)

__global__ __launch_bounds__(256) void enc_zero(float* __restrict__ p, int n) {
    int i = blockIdx.x * blockDim.x + threadIdx.x;
    if (i < n) p[i] = 0.0f;
}

__global__ __launch_bounds__(256) void enc_main(const float* __restrict__ Xg,
                                                const float* __restrict__ CW,
                                                const float* __restrict__ SC,
                                                float* __restrict__ out)
{
    extern __shared__ float smem[];
    float* XP   = smem;                      // (TILE/2) * XPLD           (99,072 B)
    float* CP   = XP  + (TILE / 2) * XPLD;   // (DD/2) * 64               (32,768 B)
    float* AP   = CP  + (DD / 2) * 64;       // (TILE/2) * APLD           (12,672 B)
    float* sqX  = AP  + (TILE / 2) * APLD;   // TILE
    float* sqC  = sqX + TILE;                // KK
    float* scl  = sqC + KK;                  // KK
    float* sums = scl + KK;                  // 256 (per-thread sumA partials)
    float* sumT = sums + 256;                // KK (reduced sum_n A)

    const int t    = threadIdx.x;
    const int w    = t >> 5;                 // wave id 0..7
    const int l    = t & 31;                 // lane id
    const int m    = l & 15;                 // lane % 16
    const int hi   = l >> 4;                 // half-wave select
    const int koff = hi * 2;                 // K-dim offset for 16x16x4 A/B operands
    const int bId  = blockIdx.y;             // batch
    const long n00 = (long)blockIdx.x * CHUNK;

    // ---- load codewords K-dim-paired: CP[d/2 * 64 + 2k + j] = C[k][d+j] ----
    for (int i = t; i < KK * DD; i += 256) {
        int k = i >> 8, d = i & 255;                  // coalesced read of C[k][d]
        CP[(d >> 1) * 64 + 2 * k + (d & 1)] = CW[i];
    }
    if (t < KK) scl[t] = SC[t];
    __syncthreads();
    if (t < KK) {
        float s = 0.0f;
        for (int dp = 0; dp < DD / 2; ++dp) {
            float c0 = CP[dp * 64 + 2 * t], c1 = CP[dp * 64 + 2 * t + 1];
            s += c0 * c0 + c1 * c1;
        }
        sqC[t] = s;
    }

    float sumA_acc = 0.0f;                   // partial of sum_n A[n, k=l]
    v8f e0 = {}, e1 = {}, e2 = {}, e3 = {};  // E accumulators: [k-tile][d-tile]
    const int D0 = w * 32;                   // this wave's D slice

    const float* Xb0 = Xg + (long)bId * DD * NB;

    for (int it = 0; it < ITERS; ++it) {
        const long n0 = n00 + (long)it * TILE;
        // ---- load X tile: global [d][n] (coalesced float4) -> LDS XP[n/2][d][2] ----
        const float* Xb = Xb0 + n0;
        for (int i = t; i < DD * (TILE / 4); i += 256) {
            int d = i / (TILE / 4), c4 = i % (TILE / 4);
            const float4 v = *(const float4*)(Xb + (long)d * NB + c4 * 4);
            int np = c4 * 2;                           // (4*c4)/2
            v2f lo = {v.x, v.y}, hl = {v.z, v.w};
            *(v2f*)(&XP[np * XPLD + 2 * d])       = lo;   // ds_store_b64
            *(v2f*)(&XP[(np + 1) * XPLD + 2 * d]) = hl;   // ds_store_b64
        }
        // ---- prefetch next tile toward L2 while this one is consumed ----
        if (it + 1 < ITERS) {
            const float* Xn = Xb0 + n0 + TILE;
            for (int i = t; i < DD * 3; i += 256) {    // 96 floats = 3 x 128B lines per d-row
                int d = i / 3, seg = i % 3;
                __builtin_prefetch(Xn + (long)d * NB + seg * 32, 0, 3);
            }
        }
        __syncthreads();

        // ---- |x|^2 per row ----
        if (t < TILE) {
            float s = 0.0f;
            const float* xr = &XP[(t >> 1) * XPLD + (t & 1)];
            for (int d = 0; d < DD; ++d) { float x = xr[2 * d]; s += x * x; }
            sqX[t] = s;
        }
        __syncthreads();

        // ---- stage A: cross[16 rows x 32 k] per wave (waves 0..5) ----
        if (w < TILE / 16) {
            const int nb = 16 * w;
            const int nr = nb + m;                     // this lane's row (A-op M=lane%16)
            v8f c0 = {}, c1 = {};
            const float* xp = &XP[(nr >> 1) * XPLD + (nr & 1) + 2 * koff];
            const float* cp = &CP[hi * 64 + 2 * m];
#pragma unroll 8
            for (int d = 0; d < DD; d += 4) {
                v2f a;  a.x = xp[0]; a.y = xp[2];              // ds_load_2addr_b32
                v2f b0 = *(const v2f*)(cp);                    // b64
                v2f b1 = *(const v2f*)(cp + 32);               // b64
                c0 = __builtin_amdgcn_wmma_f32_16x16x4_f32(false, a, false, b0, (short)0, c0, false, false);
                c1 = __builtin_amdgcn_wmma_f32_16x16x4_f32(false, a, false, b1, (short)0, c1, false, false);
                xp += 8;                                       // d += 4 (paired)
                cp += 128;                                     // dp += 2
            }
            // dist * scale -> AP   (C/D layout: VGPR v -> M = v + 8*hi, N = lane%16)
            for (int v = 0; v < 8; ++v) {
                int nl = nb + v + hi * 8;
                float sx = sqX[nl];
                int base = (nl >> 1) * APLD + (nl & 1);
                int k0 = m, k1 = 16 + m;
                AP[base + 2 * k0] = (sx - 2.0f * c0[v] + sqC[k0]) * scl[k0];
                AP[base + 2 * k1] = (sx - 2.0f * c1[v] + sqC[k1]) * scl[k1];
            }
        }
        __syncthreads();

        // ---- softmax over k (32) per row ----
        if (t < TILE) {
            float* row = &AP[(t >> 1) * APLD + (t & 1)];
            float mx = row[0];
            for (int k = 1; k < KK; ++k) mx = fmaxf(mx, row[2 * k]);
            float s = 0.0f;
            for (int k = 0; k < KK; ++k) { float e = __expf(row[2 * k] - mx); row[2 * k] = e; s += e; }
            float inv = 1.0f / s;
            for (int k = 0; k < KK; ++k) row[2 * k] *= inv;
        }
        __syncthreads();

        // ---- sum_n A partials: thread (w,l) sums rows 12w..12w+11 of column k=l ----
        {
            float s = 0.0f;
            for (int r = 12 * w; r < 12 * w + 12; ++r)
                s += AP[(r >> 1) * APLD + 2 * l + (r & 1)];
            sumA_acc += s;
        }

        // ---- stage B: E[32 x 32-slice] += A^T x Xtile over n=0..95 ----
        {
            const float* ap = &AP[hi * APLD + 2 * m];          // (nk>>1) = n/2 + hi
            const float* bp = &XP[hi * XPLD + 2 * (D0 + m)];
#pragma unroll 6
            for (int n = 0; n < TILE; n += 4) {
                v2f a0 = *(const v2f*)(ap);                    // b64
                v2f a1 = *(const v2f*)(ap + 32);               // b64
                v2f b0 = *(const v2f*)(bp);                    // b64
                v2f b1 = *(const v2f*)(bp + 32);               // b64
                e0 = __builtin_amdgcn_wmma_f32_16x16x4_f32(false, a0, false, b0, (short)0, e0, false, false);
                e1 = __builtin_amdgcn_wmma_f32_16x16x4_f32(false, a0, false, b1, (short)0, e1, false, false);
                e2 = __builtin_amdgcn_wmma_f32_16x16x4_f32(false, a1, false, b0, (short)0, e2, false, false);
                e3 = __builtin_amdgcn_wmma_f32_16x16x4_f32(false, a1, false, b1, (short)0, e3, false, false);
                ap += 2 * APLD;
                bp += 2 * XPLD;
            }
        }
        __syncthreads();   // protect XP/AP before next tile's overwrite
    }

    // ---- reduce sum_n A across waves ----
    sums[t] = sumA_acc;
    __syncthreads();
    if (t < KK) {
        float s = 0.0f;
        for (int ww = 0; ww < 8; ++ww) s += sums[ww * 32 + t];
        sumT[t] = s;
    }
    __syncthreads();

    // ---- E -= sumA[k]*C[k][d]; atomic-accumulate chunk contribution ----
    float* outB = out + (long)bId * KK * DD;
    for (int v = 0; v < 8; ++v) {
        int k0 = v + hi * 8;             // k for e0/e1 (tile kt=0)
        int k1 = 16 + k0;                // k for e2/e3 (tile kt=1)
        int d0 = D0 + m, d1 = D0 + 16 + m;
        float c00 = CP[(d0 >> 1) * 64 + 2 * k0 + (d0 & 1)];
        float c01 = CP[(d1 >> 1) * 64 + 2 * k0 + (d1 & 1)];
        float c10 = CP[(d0 >> 1) * 64 + 2 * k1 + (d0 & 1)];
        float c11 = CP[(d1 >> 1) * 64 + 2 * k1 + (d1 & 1)];
        unsafeAtomicAdd(&outB[k0 * DD + d0], e0[v] - sumT[k0] * c00);
        unsafeAtomicAdd(&outB[k0 * DD + d1], e1[v] - sumT[k0] * c01);
        unsafeAtomicAdd(&outB[k1 * DD + d0], e2[v] - sumT[k1] * c10);
        unsafeAtomicAdd(&outB[k1 * DD + d1], e3[v] - sumT[k1] * c11);
    }
}

extern "C" void kernel_launch(void* const* d_in, const int* in_sizes, int n_in,
                              void* d_out, int out_size, void* d_ws, size_t ws_size,
                              hipStream_t stream) {
    const float* X  = (const float*)d_in[0];   // [32, 256, 96, 96]
    const float* CW = (const float*)d_in[1];   // [32, 256]
    const float* SC = (const float*)d_in[2];   // [32]
    float* out = (float*)d_out;                // [32, 32, 256]

    enc_zero<<<(out_size + 255) / 256, 256, 0, stream>>>(out, out_size);

    const size_t shmem = (size_t)((TILE / 2) * XPLD + (DD / 2) * 64 + (TILE / 2) * APLD +
                                  TILE + KK + KK + 256 + KK) * sizeof(float);
    dim3 grid(NB / CHUNK, 32);   // (12, 32)
    enc_main<<<grid, 256, shmem, stream>>>(X, CW, SC, out);
}